// DigiCap_38998303047937
// MI455X (gfx1250) — compile-verified
//
#include <hip/hip_runtime.h>
#include <math.h>

// ---------------------------------------------------------------------------
// DigitCaps dynamic routing, fused single kernel for MI455X (gfx1250).
//
// inputs_hat[b,i,j,q] = sum_p inputs[b,i,p] * W[i,j,p,q]
//   => per input-capsule i: GEMM  [16b x 8p] * [8p x 160(j,q)]  via
//      V_WMMA_F32_16X16X4_F32 chained twice over K (8 = 2*4), 10 N-tiles.
//
// Key trick: s = sum_i c .* hat  is computed by folding c[b,i,j] into the A
// fragment rows (A layout has M = lane&15 in ALL lanes, which is exactly the
// lane that computed softmax for batch row b), so the weighted reduction over
// i happens inside the WMMA C-accumulator with zero cross-lane traffic.
//
// Routing (3 iters) fused; inputs_hat recomputed per sweep (188 MB if
// materialized -> recompute is free at 0.755 GFLOP vs 23.3 TB/s HBM).
// Routing logits b[i, batch, j] live in d_ws (11.8 MB), written before read.
// ---------------------------------------------------------------------------

typedef __attribute__((ext_vector_type(2))) float v2f;
typedef __attribute__((ext_vector_type(8))) float v8f;

#define BATCH 256
#define IC    1152   // input capsules
#define P     8      // input dim
#define NC    10     // output capsules
#define DV    16     // output dim
#define BT    16     // batch tile per workgroup (WMMA M)
#define NWAVE 8      // waves per workgroup

#define WMMA_F32_K4(a, b, c) \
  __builtin_amdgcn_wmma_f32_16x16x4_f32(false, (a), false, (b), (short)0, (c), false, false)

__global__ __launch_bounds__(256) void caps_route_kernel(
    const float* __restrict__ inp,   // [256,1152,8]
    const float* __restrict__ Wt,    // [1152,10,8,16]
    float* __restrict__ out,         // [256,10,16]
    float* __restrict__ ws_b)        // [1152,256,10] routing logits
{
  __shared__ float s_lds[BT][NC][DV];   // s accumulator across waves
  __shared__ float v_lds[BT][NC][DV];   // squashed outputs v

  const int tid  = threadIdx.x;
  const int wave = tid >> 5;
  const int lane = tid & 31;
  const int m    = lane & 15;   // WMMA: A row (b) / B-D column (q)
  const int hi   = lane >> 4;   // half-wave selector (K pairs / M+8 in D)
  const int bt   = blockIdx.x * BT;

  for (int t = 0; t < 3; ++t) {
    // ---- zero s accumulator ------------------------------------------------
    for (int idx = tid; idx < BT * NC * DV; idx += 256)
      ((float*)s_lds)[idx] = 0.0f;
    __syncthreads();

    // ---- sweep 1: s[b,j,q] = sum_i,p (c[b,i,j]*inp[b,i,p]) * W[i,j,p,q] ----
    v8f zero = {};
    v8f acc[NC];
#pragma unroll
    for (int j = 0; j < NC; ++j) acc[j] = zero;

    for (int i = wave; i < IC; i += NWAVE) {
      // routing coefficients c[b=m, i, j]; every lane holds row b = m, which
      // matches the A-fragment row it feeds into the WMMA.
      float cv[NC];
      if (t == 0) {
#pragma unroll
        for (int j = 0; j < NC; ++j) cv[j] = 0.1f;   // softmax(0) over 10
      } else {
        const float* bw = ws_b + ((size_t)i * BATCH + bt + m) * NC;
        float mx = -1e30f;
#pragma unroll
        for (int j = 0; j < NC; ++j) { cv[j] = bw[j]; mx = fmaxf(mx, cv[j]); }
        float ssum = 0.0f;
#pragma unroll
        for (int j = 0; j < NC; ++j) { cv[j] = __expf(cv[j] - mx); ssum += cv[j]; }
        float inv = 1.0f / ssum;
#pragma unroll
        for (int j = 0; j < NC; ++j) cv[j] *= inv;
      }

      // A fragment: 16x4 f32 layout, two K-chunks (p 0..3 and 4..7)
      const float* ap = inp + ((size_t)(bt + m) * IC + i) * P + 2 * hi;
      v2f a0 = *(const v2f*)ap;         // p = 2hi, 2hi+1
      v2f a1 = *(const v2f*)(ap + 4);   // p = 4+2hi, 5+2hi

      // stream next W block toward L0/L2
      if (i + NWAVE < IC)
        __builtin_prefetch(Wt + (size_t)(i + NWAVE) * NC * P * DV, 0, 3);

      const float* wbase = Wt + (size_t)i * NC * P * DV;
#pragma unroll
      for (int j = 0; j < NC; ++j) {
        const float* wp = wbase + (j * P + 2 * hi) * DV + m;
        v2f b0, b1;
        b0.x = wp[0];       b0.y = wp[DV];        // rows 2hi, 2hi+1
        b1.x = wp[4 * DV];  b1.y = wp[5 * DV];    // rows 4+2hi, 5+2hi
        v2f a0s = a0 * cv[j];                     // fold c into A rows
        v2f a1s = a1 * cv[j];
        acc[j] = WMMA_F32_K4(a0s, b0, acc[j]);    // accumulate s over i in C
        acc[j] = WMMA_F32_K4(a1s, b1, acc[j]);
      }
    }

    // ---- cross-wave reduction into LDS ------------------------------------
    // D: lane(hi,m), reg r -> s-partial[b=r+8hi, q=m]
#pragma unroll
    for (int j = 0; j < NC; ++j)
#pragma unroll
      for (int r = 0; r < 8; ++r)
        atomicAdd(&s_lds[r + 8 * hi][j][m], acc[j][r]);   // ds_add_f32
    __syncthreads();

    // ---- squash: v = (|s|^2/(1+|s|^2)) * s/sqrt(|s|^2+eps) -----------------
    if (tid < BT * NC) {
      int b = tid / NC, j = tid % NC;
      float sv[DV];
      float sq = 0.0f;
#pragma unroll
      for (int q = 0; q < DV; ++q) { sv[q] = s_lds[b][j][q]; sq += sv[q] * sv[q]; }
      float scale = sq / ((1.0f + sq) * sqrtf(sq + 1e-7f));
#pragma unroll
      for (int q = 0; q < DV; ++q) {
        float vq = scale * sv[q];
        if (t == 2) out[((size_t)(bt + b) * NC + j) * DV + q] = vq;
        else        v_lds[b][j][q] = vq;
      }
    }
    __syncthreads();

    // ---- sweep 2 (iters 0,1): logits b += sum_q hat[b,i,j,q]*v[b,j,q] ------
    if (t < 2) {
      for (int i = wave; i < IC; i += NWAVE) {
        const float* ap = inp + ((size_t)(bt + m) * IC + i) * P + 2 * hi;
        v2f a0 = *(const v2f*)ap;
        v2f a1 = *(const v2f*)(ap + 4);
        const float* wbase = Wt + (size_t)i * NC * P * DV;
        float binc[8];
#pragma unroll
        for (int r = 0; r < 8; ++r) binc[r] = 0.0f;
#pragma unroll
        for (int j = 0; j < NC; ++j) {
          const float* wp = wbase + (j * P + 2 * hi) * DV + m;
          v2f b0, b1;
          b0.x = wp[0];       b0.y = wp[DV];
          b1.x = wp[4 * DV];  b1.y = wp[5 * DV];
          v8f h = {};
          h = WMMA_F32_K4(a0, b0, h);             // raw hat tile for capsule j
          h = WMMA_F32_K4(a1, b1, h);
#pragma unroll
          for (int r = 0; r < 8; ++r) {
            float prod = h[r] * v_lds[r + 8 * hi][j][m];
            // reduce over q (= lanes within each half-wave of 16)
            prod += __shfl_xor(prod, 1, 32);
            prod += __shfl_xor(prod, 2, 32);
            prod += __shfl_xor(prod, 4, 32);
            prod += __shfl_xor(prod, 8, 32);
            if (m == j) binc[r] = prod;   // lane m==j keeps agreement for j
          }
        }
        if (m < NC) {
#pragma unroll
          for (int r = 0; r < 8; ++r) {
            size_t idx = ((size_t)i * BATCH + bt + r + 8 * hi) * NC + m;
            if (t == 0) ws_b[idx] = binc[r];          // b starts at 0
            else        ws_b[idx] = ws_b[idx] + binc[r];
          }
        }
      }
    }
    __syncthreads();
  }
}

extern "C" void kernel_launch(void* const* d_in, const int* in_sizes, int n_in,
                              void* d_out, int out_size, void* d_ws, size_t ws_size,
                              hipStream_t stream) {
  const float* inp = (const float*)d_in[0];   // [256,1152,8] fp32
  const float* Wt  = (const float*)d_in[1];   // [1152,10,8,16] fp32
  float* out       = (float*)d_out;           // [256,10,16] fp32
  float* ws_b      = (float*)d_ws;            // routing logits, 11.8 MB

  caps_route_kernel<<<dim3(BATCH / BT), dim3(256), 0, stream>>>(inp, Wt, out, ws_b);
}